// TextToSpatialRetrievalHead_20693152432925
// MI455X (gfx1250) — compile-verified
//
#include <hip/hip_runtime.h>
#include <hip/hip_bf16.h>
#include <math.h>

// ---------------------------------------------------------------------------
// TextToSpatialRetrievalHead for MI455X (gfx1250)
//   scores[64,500000] = normalize(text) @ normalize(spots)^T   (exact f32)
//   indices[64,10]    = top-10 per row (lowest index wins ties)
//
// GEMM: V_WMMA_F32_16X16X4_F32 with TWO interleaved K-accumulation chains
// (breaks the serial D->C dependency on the matrix pipe). A (normalized text,
// 64x512) resident in LDS; spot tiles streamed with
// global_load_async_to_lds_b128 + s_wait_asynccnt, double-buffered.
// Spot norms computed from LDS and applied as a column scale.
// ---------------------------------------------------------------------------

typedef __attribute__((ext_vector_type(2))) float v2f;
typedef __attribute__((ext_vector_type(8))) float v8f;

#define NTOT   500000      // spots
#define DIM    512
#define MROWS  64
#define NB     32          // spots per tile (500000 = 32 * 15625, exact)
#define LDA    516         // padded LDS row stride (floats): +4 dwords kills bank conflicts
#define CHUNK  8192        // top-k stage-1 chunk

__device__ __forceinline__ void async_ld_b128(unsigned lds_byte_off, const void* gaddr) {
    // GV mode: vdst = LDS byte address, vaddr = 64-bit global address, saddr = off
    asm volatile("global_load_async_to_lds_b128 %0, %1, off"
                 :: "v"(lds_byte_off), "v"(gaddr) : "memory");
}
__device__ __forceinline__ void wait_async0() {
    asm volatile("s_wait_asynccnt 0" ::: "memory");
}
__device__ __forceinline__ unsigned lds_off(const void* p) {
    // generic pointer to LDS: low 32 bits are the wave-relative DS byte offset
    return (unsigned)(uintptr_t)p;
}

// ---------------------------------------------------------------------------
// Kernel 1: L2-normalize text rows (matches F.normalize: x / max(||x||, eps))
// ---------------------------------------------------------------------------
__global__ void k_norm_text(const float* __restrict__ x, float* __restrict__ y) {
    int row = blockIdx.x, tid = threadIdx.x;            // 64 blocks x 256 thr
    const float2* xr = (const float2*)(x + (size_t)row * DIM);
    float2 v = xr[tid];
    float ss = v.x * v.x + v.y * v.y;
#pragma unroll
    for (int o = 16; o > 0; o >>= 1) ss += __shfl_down(ss, o, 32);
    __shared__ float part[8];
    __shared__ float scale;
    if ((tid & 31) == 0) part[tid >> 5] = ss;
    __syncthreads();
    if (tid == 0) {
        float t = 0.f;
        for (int w = 0; w < 8; ++w) t += part[w];
        scale = 1.0f / fmaxf(sqrtf(t), 1e-12f);
    }
    __syncthreads();
    float2 o; o.x = v.x * scale; o.y = v.y * scale;
    ((float2*)(y + (size_t)row * DIM))[tid] = o;
}

// ---------------------------------------------------------------------------
// Kernel 2: WMMA GEMM + fused spot normalization.
//   block = 256 thr (8 waves). LDS: A[64][516] + 2x B[32][516] + invn[2][32].
//   Wave w -> C tile (m = 16*(w&3), n = 16*(w>>2)) of the 64x32 block output.
// ---------------------------------------------------------------------------
__global__ void k_gemm(const float* __restrict__ tnorm,
                       const float* __restrict__ spots,
                       float* __restrict__ scores, int ntiles) {
    extern __shared__ float smem[];
    float* As   = smem;                        // 64*516
    float* Bs0  = As  + MROWS * LDA;           // 32*516
    float* Bs1  = Bs0 + NB * LDA;              // 32*516
    float* invn = Bs1 + NB * LDA;              // 2*32

    const int tid = threadIdx.x;

    // --- async-load A (64 x 512 f32 = 8192 x b128), once per block ---
    {
        unsigned base = lds_off(As);
        for (int i = tid; i < MROWS * (DIM / 4); i += 256) {
            int r = i >> 7, c = i & 127;      // c indexes 16B units
            async_ld_b128(base + (unsigned)(r * LDA + c * 4) * 4,
                          tnorm + (size_t)r * DIM + c * 4);
        }
    }

    auto loadB = [&](int buf, int tile) {
        unsigned base = lds_off(buf ? Bs1 : Bs0);
        const float* g = spots + (size_t)tile * NB * DIM;
        for (int i = tid; i < NB * (DIM / 4); i += 256) {
            int r = i >> 7, c = i & 127;
            async_ld_b128(base + (unsigned)(r * LDA + c * 4) * 4,
                          g + (size_t)r * DIM + c * 4);
        }
    };

    int t = blockIdx.x;
    if (t >= ntiles) return;
    int buf = 0;
    loadB(0, t);

    // WMMA fragment addressing (ISA 7.12.2):
    //  A 16x4 f32: lanes 0-15 hold M=lane,(K, K+1); lanes 16-31 hold (K+2, K+3)
    //  C/D 16x16:  lanes 0-15 N=lane, VGPR i -> M=i; lanes 16-31 -> M=i+8
    const int lane = tid & 31, wv = tid >> 5;
    const int mt = wv & 3, nt = wv >> 2;       // m tile 0..3, n tile 0..1
    const int lh = lane & 15, hi = lane >> 4;
    const float* Arow = As + (mt * 16 + lh) * LDA + 2 * hi;

    while (t < ntiles) {
        const int tn = t + gridDim.x;

        wait_async0();
        __syncthreads();                        // current B tile (and A) ready

        // per-spot inverse norms from LDS: 8 threads per row, shfl-reduce
        {
            const float* bb = buf ? Bs1 : Bs0;
            int r = tid >> 3, sub = tid & 7;
            const float* br = bb + r * LDA + sub * 64;
            float ss = 0.f;
#pragma unroll 8
            for (int j = 0; j < 64; ++j) ss += br[j] * br[j];
            ss += __shfl_xor(ss, 1, 32);
            ss += __shfl_xor(ss, 2, 32);
            ss += __shfl_xor(ss, 4, 32);
            if (sub == 0) invn[buf * NB + r] = 1.0f / fmaxf(sqrtf(ss), 1e-12f);
        }

        if (tn < ntiles) loadB(buf ^ 1, tn);    // prefetch next tile (overlaps compute)
        __syncthreads();                        // invn visible

        const float* Brow = (buf ? Bs1 : Bs0) + (nt * 16 + lh) * LDA + 2 * hi;

        // Two independent accumulation chains over interleaved K halves:
        // acc0 <- k, acc1 <- k+4 (step 8). Breaks WMMA D->C serialization.
        v8f acc0 = {};
        v8f acc1 = {};
#pragma unroll 4
        for (int k = 0; k < DIM; k += 8) {
            v2f a0 = *(const v2f*)(Arow + k);        // ds_load_b64
            v2f b0 = *(const v2f*)(Brow + k);        // ds_load_b64
            v2f a1 = *(const v2f*)(Arow + k + 4);
            v2f b1 = *(const v2f*)(Brow + k + 4);
            acc0 = __builtin_amdgcn_wmma_f32_16x16x4_f32(
                false, a0, false, b0, (short)0, acc0, false, false);
            acc1 = __builtin_amdgcn_wmma_f32_16x16x4_f32(
                false, a1, false, b1, (short)0, acc1, false, false);
        }

        // scale columns by spot inv-norm, write scores[m, n_global]
        const float sc = invn[buf * NB + nt * 16 + lh];
        const size_t ncol = (size_t)t * NB + nt * 16 + lh;
        float* orow = scores + (size_t)(mt * 16 + 8 * hi) * NTOT + ncol;
#pragma unroll
        for (int i = 0; i < 8; ++i) orow[(size_t)i * NTOT] = (acc0[i] + acc1[i]) * sc;

        buf ^= 1;
        t = tn;
    }
}

// ---------------------------------------------------------------------------
// Kernel 3: top-k stage 1 — per (row, chunk) find top-10 candidates.
// Iterative arg-max with smaller-index tie preference (matches lax.top_k).
// ---------------------------------------------------------------------------
__global__ void k_topk1(const float* __restrict__ scores,
                        float2* __restrict__ cand, int nchunks) {
    const int row = blockIdx.y, ch = blockIdx.x, tid = threadIdx.x;
    __shared__ float vals[CHUNK];               // 32 KB
    __shared__ float rv[8];
    __shared__ int   ri[8];

    const float* src = scores + (size_t)row * NTOT + (size_t)ch * CHUNK;
    const int limit = min(CHUNK, NTOT - ch * CHUNK);
    for (int i = tid; i < CHUNK; i += 256)
        vals[i] = (i < limit) ? src[i] : -__builtin_inff();
    __syncthreads();

    for (int sel = 0; sel < 10; ++sel) {
        float bv = -__builtin_inff(); int bi = 0x7fffffff;
        for (int i = tid; i < CHUNK; i += 256) {
            float v = vals[i];
            if (v > bv || (v == bv && i < bi)) { bv = v; bi = i; }
        }
#pragma unroll
        for (int o = 16; o > 0; o >>= 1) {
            float ov = __shfl_down(bv, o, 32);
            int   oi = __shfl_down(bi, o, 32);
            if (ov > bv || (ov == bv && oi < bi)) { bv = ov; bi = oi; }
        }
        if ((tid & 31) == 0) { rv[tid >> 5] = bv; ri[tid >> 5] = bi; }
        __syncthreads();
        if (tid == 0) {
            float fv = rv[0]; int fi = ri[0];
            for (int w = 1; w < 8; ++w)
                if (rv[w] > fv || (rv[w] == fv && ri[w] < fi)) { fv = rv[w]; fi = ri[w]; }
            cand[((size_t)row * nchunks + ch) * 10 + sel] =
                make_float2(fv, __int_as_float(ch * CHUNK + fi));
            vals[fi] = -__builtin_inff();
        }
        __syncthreads();
    }
}

// ---------------------------------------------------------------------------
// Kernel 4: top-k stage 2 — merge 62*10 candidates per row, emit indices.
// Indices stored as float values (exact: < 2^24) into tail of d_out.
// ---------------------------------------------------------------------------
__global__ void k_topk2(const float2* __restrict__ cand,
                        float* __restrict__ out_idx, int nchunks) {
    const int row = blockIdx.x, tid = threadIdx.x;
    const int n = nchunks * 10;                  // 620
    __shared__ float v[640];
    __shared__ int   ix[640];
    __shared__ float rv[8]; __shared__ int ri[8]; __shared__ int rp[8];

    for (int i = tid; i < n; i += 256) {
        float2 c = cand[(size_t)row * n + i];
        v[i] = c.x; ix[i] = __float_as_int(c.y);
    }
    __syncthreads();

    for (int sel = 0; sel < 10; ++sel) {
        float bv = -__builtin_inff(); int bi = 0x7fffffff; int bp = -1;
        for (int i = tid; i < n; i += 256) {
            if (v[i] > bv || (v[i] == bv && ix[i] < bi)) { bv = v[i]; bi = ix[i]; bp = i; }
        }
#pragma unroll
        for (int o = 16; o > 0; o >>= 1) {
            float ov = __shfl_down(bv, o, 32);
            int   oi = __shfl_down(bi, o, 32);
            int   op = __shfl_down(bp, o, 32);
            if (ov > bv || (ov == bv && oi < bi)) { bv = ov; bi = oi; bp = op; }
        }
        if ((tid & 31) == 0) { rv[tid >> 5] = bv; ri[tid >> 5] = bi; rp[tid >> 5] = bp; }
        __syncthreads();
        if (tid == 0) {
            float fv = rv[0]; int fi = ri[0]; int fp = rp[0];
            for (int w = 1; w < 8; ++w)
                if (rv[w] > fv || (rv[w] == fv && ri[w] < fi)) { fv = rv[w]; fi = ri[w]; fp = rp[w]; }
            out_idx[(size_t)row * 10 + sel] = (float)fi;
            if (fp >= 0) v[fp] = -__builtin_inff();
        }
        __syncthreads();
    }
}

// ---------------------------------------------------------------------------
extern "C" void kernel_launch(void* const* d_in, const int* in_sizes, int n_in,
                              void* d_out, int out_size, void* d_ws, size_t ws_size,
                              hipStream_t stream) {
    const float* text  = (const float*)d_in[0];   // 64*512
    const float* spots = (const float*)d_in[1];   // 500000*512
    // d_in[2] = top_k (fixed 10 in reference)

    float* scores  = (float*)d_out;                        // 64*500000
    float* out_idx = scores + (size_t)MROWS * NTOT;        // 64*10 (as floats)

    float*  tnorm = (float*)d_ws;                          // 128 KB
    float2* cand  = (float2*)((char*)d_ws + (size_t)MROWS * DIM * sizeof(float));
    // workspace use: 128 KB + 64*62*10*8 B = ~448 KB total

    k_norm_text<<<MROWS, 256, 0, stream>>>(text, tnorm);

    const int ntiles = NTOT / NB;                          // 15625 (exact)
    const size_t smem = (size_t)(MROWS * LDA + 2 * NB * LDA + 2 * NB) * sizeof(float);
    int gemm_blocks = 2048;
    if (gemm_blocks > ntiles) gemm_blocks = ntiles;
    k_gemm<<<gemm_blocks, 256, smem, stream>>>(tnorm, spots, scores, ntiles);

    const int nchunks = (NTOT + CHUNK - 1) / CHUNK;        // 62
    dim3 g1(nchunks, MROWS);
    k_topk1<<<g1, 256, 0, stream>>>(scores, cand, nchunks);
    k_topk2<<<MROWS, 256, 0, stream>>>(cand, out_idx, nchunks);
}